// MorphologicalConvolutionR2_10101763080231
// MI455X (gfx1250) — compile-verified
//
#include <hip/hip_runtime.h>
#include <cstdint>
#include <cstddef>

// Problem constants (from reference): B=16, C=96, H=W=224, KH=KW=11
#define CC   96
#define HH   224
#define WW   224
#define KH   11
#define KW   11
#define HALO 5

// Tiling: exact (224 = 4*56 = 7*32), no partial tiles.
#define TILE_W 56
#define TILE_H 32
#define RH     4                       // rows per thread
#define RW     2                       // adjacent cols per thread (float2 pairs)
#define IN_W   (TILE_W + 2 * HALO)     // 66 (even -> 8B-aligned rows)
#define IN_H   (TILE_H + 2 * HALO)     // 42
#define NTHREADS ((TILE_W / RW) * (TILE_H / RH))  // 28 * 8 = 224 = 7 waves

#define AS1 __attribute__((address_space(1)))
#define AS3 __attribute__((address_space(3)))

typedef __attribute__((ext_vector_type(2))) float v2f;

// Packed f32 add (VOP3P): d = a + k, two f32 lanes per instruction.
// k is wave-uniform -> 64-bit SGPR pair source (single constant-bus read).
// op_sel/op_sel_hi must be explicitly [0,0] on gfx1250 packed-f32 ops.
static __device__ __forceinline__ v2f pk_add_vs(v2f a, v2f k)
{
    v2f d;
    asm("v_pk_add_f32 %0, %1, %2 op_sel:[0,0] op_sel_hi:[0,0]"
        : "=v"(d) : "v"(a), "s"(k));
    return d;
}

__global__ __launch_bounds__(NTHREADS) void
morph_minplus_11x11(const float* __restrict__ x,
                    const float* __restrict__ kern,
                    float* __restrict__ out)
{
    // Two copies of the input tile: B is shifted left by one column, so the
    // column pair (col+j, col+j+1) is an ALIGNED 64-bit LDS load for every j
    // (copy A for even j, copy B for odd j) -> ds_load_b64 into an even VGPR
    // pair feeding v_pk_add_f32 with zero register-pair copies.
    __shared__ float stileA[IN_H * IN_W];
    __shared__ float stileB[IN_H * IN_W];

    const int tid = threadIdx.x;
    const int tw0 = blockIdx.x * TILE_W;
    const int th0 = blockIdx.y * TILE_H;
    const int bc  = blockIdx.z;              // b*C + c
    const int c   = bc % CC;

    const float* __restrict__ plane = x + (size_t)bc * (HH * WW);
    const float* __restrict__ kq    = kern + (size_t)c * (KH * KW);  // wave-uniform

    const float INF = __builtin_inff();

    // ---- Fill both LDS tiles (with +inf out-of-bounds halo) ----
    // CDNA5 async global->LDS path (ASYNCcnt).
    for (int e = tid; e < IN_H * IN_W; e += NTHREADS) {
        const int r  = e / IN_W;
        const int cc = e - r * IN_W;
        const int gr  = th0 + r  - HALO;
        const int gcA = tw0 + cc - HALO;       // copy A: orig column cc
        const int gcB = gcA + 1;               // copy B: orig column cc+1
        const bool inr  = ((unsigned)gr  < (unsigned)HH);
        const bool inbA = inr && ((unsigned)gcA < (unsigned)WW);
        const bool inbB = inr && ((unsigned)gcB < (unsigned)WW);
#if __has_builtin(__builtin_amdgcn_global_load_async_to_lds_b32)
        if (inbA) {
            __builtin_amdgcn_global_load_async_to_lds_b32(
                (AS1 int*)(plane + (size_t)gr * WW + gcA),
                (AS3 int*)&stileA[e], 0, 0);
        } else {
            stileA[e] = INF;
        }
        if (inbB) {
            __builtin_amdgcn_global_load_async_to_lds_b32(
                (AS1 int*)(plane + (size_t)gr * WW + gcB),
                (AS3 int*)&stileB[e], 0, 0);
        } else {
            stileB[e] = INF;
        }
#else
        stileA[e] = inbA ? plane[(size_t)gr * WW + gcA] : INF;
        stileB[e] = inbB ? plane[(size_t)gr * WW + gcB] : INF;
#endif
    }
#if __has_builtin(__builtin_amdgcn_global_load_async_to_lds_b32)
#  if __has_builtin(__builtin_amdgcn_s_wait_asynccnt)
    __builtin_amdgcn_s_wait_asynccnt(0);
#  else
    asm volatile("s_wait_asynccnt 0" ::: "memory");
#  endif
#endif
    __syncthreads();

    // ---- Compute: each thread owns a 4-row x 2-col output block ----
    const int cp   = tid % (TILE_W / RW);    // column pair index, 0..27
    const int rg   = tid / (TILE_W / RW);    // row group, 0..7
    const int col  = cp * RW;                // even column
    const int row0 = rg * RH;

    v2f acc2[RH];
#pragma unroll
    for (int r = 0; r < RH; ++r) acc2[r] = (v2f){INF, INF};

#pragma unroll 1
    for (int j = 0; j < KW; ++j) {
        // 11 wave-uniform kernel taps for this column -> SGPR (k,k) pairs
        v2f kv2[KH];
#pragma unroll
        for (int i = 0; i < KH; ++i) {
            const float kv = kq[i * KW + j];
            kv2[i] = (v2f){kv, kv};
        }

        // Aligned 64-bit window loads: copy A for even j, copy B for odd j.
        const float* sp = (j & 1) ? stileB : stileA;    // uniform select
        const int jc = j & ~1;                          // even dword index
        const int lbase = row0 * IN_W + col + jc;

        v2f xv2[RH + KH - 1];
#pragma unroll
        for (int t = 0; t < RH + KH - 1; ++t)
            xv2[t] = *(const v2f*)&sp[lbase + t * IN_W];

        // Pairs of taps: v_pk_add_f32 for the adds, v_min3 for the mins.
#pragma unroll
        for (int i = 0; i + 1 < KH; i += 2) {
#pragma unroll
            for (int r = 0; r < RH; ++r) {
                const v2f a = pk_add_vs(xv2[i + r],     kv2[i]);
                const v2f b = pk_add_vs(xv2[i + 1 + r], kv2[i + 1]);
                acc2[r].x = fminf(acc2[r].x, fminf(a.x, b.x));  // v_min3
                acc2[r].y = fminf(acc2[r].y, fminf(a.y, b.y));  // v_min3
            }
        }
        // Tail tap i = KH-1
        {
#pragma unroll
            for (int r = 0; r < RH; ++r) {
                const v2f a = pk_add_vs(xv2[KH - 1 + r], kv2[KH - 1]);
                acc2[r].x = fminf(acc2[r].x, a.x);
                acc2[r].y = fminf(acc2[r].y, a.y);
            }
        }
    }

    // ---- Coalesced 64-bit stores (adjacent columns per thread) ----
    float* __restrict__ oplane = out + (size_t)bc * (HH * WW);
#pragma unroll
    for (int r = 0; r < RH; ++r) {
        *(v2f*)&oplane[(size_t)(th0 + row0 + r) * WW + (tw0 + col)] = acc2[r];
    }
}

extern "C" void kernel_launch(void* const* d_in, const int* in_sizes, int n_in,
                              void* d_out, int out_size, void* d_ws, size_t ws_size,
                              hipStream_t stream)
{
    (void)in_sizes; (void)n_in; (void)out_size; (void)d_ws; (void)ws_size;
    const float* x = (const float*)d_in[0];
    const float* k = (const float*)d_in[1];
    float* out     = (float*)d_out;

    dim3 grid(WW / TILE_W, HH / TILE_H, 16 * CC);   // (4, 7, 1536)
    morph_minplus_11x11<<<grid, dim3(NTHREADS), 0, stream>>>(x, k, out);
}